// BiMamba_44049184588220
// MI455X (gfx1250) — compile-verified
//
#include <hip/hip_runtime.h>
#include <hip/hip_bf16.h>

// ---------------------------------------------------------------------------
// BiMamba (B=2, L=1024, d_model=1024, d_inner=2048, N=16, dt_rank=64)
// fp32 end-to-end. GEMMs via V_WMMA_F32_16X16X4_F32 with DOUBLE-BUFFERED
// Tensor Data Mover staging (tensor_load_to_lds + s_wait_tensorcnt): the TDM
// prefetches chunk i+1 while the block computes chunk i. Each wave owns two
// 16x16 C tiles (block tile 32x128), reusing every A fragment across two
// WMMAs. Problem is memory-bound (~300MB @ 23.3TB/s >> 63 GFLOP f32 WMMA),
// so fp32 matrix ops + copy/compute overlap is the optimal operating point.
// ---------------------------------------------------------------------------

#define BSZ      2
#define SEQ      1024
#define DMODEL   1024
#define DINNER   2048
#define DSTATE   16
#define DTRANK   64
#define BL       (BSZ * SEQ)          // 2048 tokens

typedef float v2f  __attribute__((ext_vector_type(2)));
typedef float v8f  __attribute__((ext_vector_type(8)));
typedef unsigned int u32x4 __attribute__((ext_vector_type(4)));
typedef int   i32x4 __attribute__((ext_vector_type(4)));
typedef int   i32x8 __attribute__((ext_vector_type(8)));

#if defined(__has_builtin)
#if __has_builtin(__builtin_amdgcn_tensor_load_to_lds)
#define HAVE_TDM 1
#else
#define HAVE_TDM 0
#endif
#else
#define HAVE_TDM 0
#endif

__device__ __forceinline__ unsigned lds_off(const void* p) {
  // Low 32 bits of a generic pointer into the LDS aperture are the LDS byte
  // offset (flat addressing).
  return (unsigned)(unsigned long long)p;
}

__device__ __forceinline__ void wait_tensor0() {
#if defined(__has_builtin) && __has_builtin(__builtin_amdgcn_s_wait_tensorcnt)
  __builtin_amdgcn_s_wait_tensorcnt((short)0);
#else
  asm volatile("s_wait_tensorcnt 0x0" ::: "memory");
#endif
}

#if HAVE_TDM
// 2D tile load: tile (tile1 rows x tile0 elems), row stride stride0 (elems),
// 4-byte elements, into contiguous LDS at lds_byte_off. Rows/cols beyond
// (tdim1, tdim0) read as zero (TDM OOB rule).
__device__ __forceinline__ void tdm_load_2d(unsigned lds_byte_off,
                                            const float* gptr,
                                            unsigned tile0, unsigned tile1,
                                            unsigned tdim0, unsigned tdim1,
                                            unsigned long long stride0) {
  unsigned long long ga = (unsigned long long)gptr;
  u32x4 g0;
  i32x8 g1;
  i32x4 g2 = {0, 0, 0, 0};
  i32x4 g3 = {0, 0, 0, 0};
  g0[0] = 1u;                                        // count=1 (valid), user mode
  g0[1] = lds_byte_off;                              // lds_addr
  g0[2] = (unsigned)(ga & 0xFFFFFFFFull);            // global_addr[31:0]
  g0[3] = (unsigned)((ga >> 32) & 0x1FFFFFFull)      // global_addr[56:32]
        | (2u << 30);                                // type = 2 ("image")
  g1[0] = (int)(2u << 16);                           // data_size=2 (4B); wg_mask=0
  g1[1] = (int)((tdim0 & 0xFFFFu) << 16);            // tensor_dim0[15:0]
  g1[2] = (int)((tdim0 >> 16) | ((tdim1 & 0xFFFFu) << 16));
  g1[3] = (int)((tdim1 >> 16) | ((tile0 & 0xFFFFu) << 16));   // tile_dim0
  g1[4] = (int)(tile1 & 0xFFFFu);                    // tile_dim1 (tile_dim2=0)
  g1[5] = (int)(stride0 & 0xFFFFFFFFull);            // tensor_dim0_stride[31:0]
  g1[6] = (int)((stride0 >> 32) & 0xFFFFull);        // stride0[47:32]; stride1=0
  g1[7] = 0;
#if defined(__clang_major__) && (__clang_major__ >= 23)
  i32x8 g4 = {0, 0, 0, 0, 0, 0, 0, 0};
  __builtin_amdgcn_tensor_load_to_lds(g0, g1, g2, g3, g4, 0);
#else
  __builtin_amdgcn_tensor_load_to_lds(g0, g1, g2, g3, 0);
#endif
}
#endif

// ---------------------------------------------------------------------------
// Generic fp32 WMMA GEMM:  C[M x N] = A[M x K] * W[N x K]^T
//   A row-major (lda), W row-major (ldw) with Nw valid rows (rows >= Nw -> 0),
//   C row-major (ldc), stored at column offset col0; logical N bound-checked.
// Block: 256 threads (8 wave32), tile 32(M) x 128(N), K staged 32 at a time
// into double-buffered LDS by the TDM. Wave w -> (wm = w>>2, wn = w&3) owns
// TWO 16x16 C tiles (N and N+64): 16 wmma per K-chunk, A fragment reused.
// ---------------------------------------------------------------------------
#define BM  32
#define BNW 128
#define BK  32

__global__ __launch_bounds__(256) void wmma_gemm_xwt(
    const float* __restrict__ A, int lda,
    const float* __restrict__ W, int ldw, int Nw,
    float* __restrict__ C, int ldc, int col0,
    int M, int N, int K) {
  __shared__ float As[2][BM][BK];    //  8 KB
  __shared__ float Ws[2][BNW][BK];   // 32 KB

  const int tid   = threadIdx.x;
  const int lane  = tid & 31;
  const int wave  = tid >> 5;
  const int wm    = wave >> 2;     // 0..1
  const int wn    = wave & 3;      // 0..3
  const int khalf = lane >> 4;     // 0..1
  const int mn    = lane & 15;     // row (A) / col (B) within 16
  const int m0    = blockIdx.y * BM;
  const int n0    = blockIdx.x * BNW;
  const int nchunks = K / BK;      // K is always a multiple of 32 here

  v8f acc0 = {0.f, 0.f, 0.f, 0.f, 0.f, 0.f, 0.f, 0.f};
  v8f acc1 = {0.f, 0.f, 0.f, 0.f, 0.f, 0.f, 0.f, 0.f};

#if HAVE_TDM
  int wrows = Nw - n0;
  if (wrows > BNW) wrows = BNW;
  if (wrows < 0)   wrows = 0;
  // Prologue: stage chunk 0 into buffer 0.
  if (wave == 0) {
    tdm_load_2d(lds_off(&As[0][0][0]), A + (size_t)m0 * lda,
                BK, BM, BK, BM, (unsigned long long)lda);
    tdm_load_2d(lds_off(&Ws[0][0][0]), W + (size_t)n0 * ldw,
                BK, BNW, BK, (unsigned)wrows, (unsigned long long)ldw);
    wait_tensor0();
  }
  __syncthreads();

  for (int i = 0; i < nchunks; ++i) {
    const int buf = i & 1;
    // Prefetch chunk i+1 into the other buffer; overlaps with compute below.
    // Safe: buf^1 was last read in iteration i-1, fenced by that barrier.
    if (wave == 0 && (i + 1) < nchunks) {
      const int kn = (i + 1) * BK;
      tdm_load_2d(lds_off(&As[buf ^ 1][0][0]), A + (size_t)m0 * lda + kn,
                  BK, BM, BK, BM, (unsigned long long)lda);
      tdm_load_2d(lds_off(&Ws[buf ^ 1][0][0]), W + (size_t)n0 * ldw + kn,
                  BK, BNW, BK, (unsigned)wrows, (unsigned long long)ldw);
    }
#pragma unroll
    for (int s = 0; s < 8; ++s) {
      const int k = s * 4 + khalf * 2;
      v2f a, b0, b1;
      a.x  = As[buf][wm * 16 + mn][k];
      a.y  = As[buf][wm * 16 + mn][k + 1];
      b0.x = Ws[buf][wn * 16 + mn][k];
      b0.y = Ws[buf][wn * 16 + mn][k + 1];
      b1.x = Ws[buf][64 + wn * 16 + mn][k];
      b1.y = Ws[buf][64 + wn * 16 + mn][k + 1];
      acc0 = __builtin_amdgcn_wmma_f32_16x16x4_f32(
          false, a, false, b0, (short)0, acc0, false, false);
      acc1 = __builtin_amdgcn_wmma_f32_16x16x4_f32(
          false, a, false, b1, (short)0, acc1, false, false);
    }
    // Wave 0 waits for the prefetch to land; barrier publishes it to all.
    if (wave == 0) wait_tensor0();
    __syncthreads();
  }
#else
  // Manual float4 staging fallback (single-buffered).
  for (int i = 0; i < nchunks; ++i) {
    const int k0 = i * BK;
    __syncthreads();
    {
      int r  = tid >> 3;             // 0..31
      int c4 = (tid & 7) * 4;        // 0..28
      const float* src = A + (size_t)(m0 + r) * lda + k0 + c4;
      *(float4*)&As[0][r][c4] = *(const float4*)src;
    }
    {
      int r  = tid >> 1;             // 0..127
      int c16 = (tid & 1) * 16;      // 0 or 16
      if (n0 + r < Nw) {
        const float* src = W + (size_t)(n0 + r) * ldw + k0 + c16;
#pragma unroll
        for (int q = 0; q < 4; ++q)
          *(float4*)&Ws[0][r][c16 + 4 * q] = *(const float4*)(src + 4 * q);
      } else {
        float4 z = {0.f, 0.f, 0.f, 0.f};
#pragma unroll
        for (int q = 0; q < 4; ++q)
          *(float4*)&Ws[0][r][c16 + 4 * q] = z;
      }
    }
    __syncthreads();
#pragma unroll
    for (int s = 0; s < 8; ++s) {
      const int k = s * 4 + khalf * 2;
      v2f a, b0, b1;
      a.x  = As[0][wm * 16 + mn][k];
      a.y  = As[0][wm * 16 + mn][k + 1];
      b0.x = Ws[0][wn * 16 + mn][k];
      b0.y = Ws[0][wn * 16 + mn][k + 1];
      b1.x = Ws[0][64 + wn * 16 + mn][k];
      b1.y = Ws[0][64 + wn * 16 + mn][k + 1];
      acc0 = __builtin_amdgcn_wmma_f32_16x16x4_f32(
          false, a, false, b0, (short)0, acc0, false, false);
      acc1 = __builtin_amdgcn_wmma_f32_16x16x4_f32(
          false, a, false, b1, (short)0, acc1, false, false);
    }
  }
#endif

  // C/D layout: lane 0-15 N=lane, VGPR r -> M=r (lanes<16) / M=r+8 (lanes>=16)
  const int nlog0 = n0 + wn * 16 + mn;
  const int nlog1 = nlog0 + 64;
  if (nlog0 < N) {
    const int col = col0 + nlog0;
#pragma unroll
    for (int r = 0; r < 8; ++r) {
      const int row = m0 + wm * 16 + khalf * 8 + r;
      C[(size_t)row * ldc + col] = acc0[r];
    }
  }
  if (nlog1 < N) {
    const int col = col0 + nlog1;
#pragma unroll
    for (int r = 0; r < 8; ++r) {
      const int row = m0 + wm * 16 + khalf * 8 + r;
      C[(size_t)row * ldc + col] = acc1[r];
    }
  }
}

// ---------------------------------------------------------------------------
// Flip along L within each batch:
//   dst[(b, L-1-l), col0 + c] = src[(b, l), c]
// ---------------------------------------------------------------------------
__global__ __launch_bounds__(256) void flip_l_kernel(
    const float* __restrict__ src, float* __restrict__ dst,
    int C, int ld_src, int ld_dst, int col0) {
  int idx = blockIdx.x * 256 + threadIdx.x;
  if (idx >= BL * C) return;
  int c = idx % C;
  int t = idx / C;
  int l = t & (SEQ - 1);
  int b = t >> 10;
  int trev = (b << 10) + (SEQ - 1 - l);
  dst[(size_t)trev * ld_dst + col0 + c] = src[(size_t)t * ld_src + c];
}

// ---------------------------------------------------------------------------
// Depthwise causal conv (d_conv=4) + bias + SiLU.
//   xz: (BL, 4096), xi = cols [0, 2048). Output xc: (BL, 2048).
// ---------------------------------------------------------------------------
__global__ __launch_bounds__(256) void conv_silu_kernel(
    const float* __restrict__ xz, const float* __restrict__ cw,
    const float* __restrict__ cb, float* __restrict__ xc) {
  int idx = blockIdx.x * 256 + threadIdx.x;
  if (idx >= BL * DINNER) return;
  int e = idx & (DINNER - 1);
  int t = idx >> 11;
  int l = t & (SEQ - 1);
  int b = t >> 10;
  float acc = cb[e];
#pragma unroll
  for (int k = 0; k < 4; ++k) {
    int ls = l + k - 3;
    if (ls >= 0)
      acc += cw[e * 4 + k] * xz[(size_t)((b << 10) + ls) * (2 * DINNER) + e];
  }
  float s = acc / (1.f + __expf(-acc));   // SiLU
  xc[(size_t)t * DINNER + e] = s;
}

// ---------------------------------------------------------------------------
// Selective scan: one lane per (batch, channel e). 16-state recurrence in
// registers, softplus(delta), + skip x*D and SiLU(z) gate fused.
//   delta_pre: (BL, 2048)   pre-softplus dt projection
//   dbl:       (BL, 128)    cols 64..79 = B_t, cols 80..95 = C_t
//   xz:        (BL, 4096)   z = cols [2048, 4096)
//   yg out:    (BL, 2048)
// ---------------------------------------------------------------------------
__global__ __launch_bounds__(256) void scan_kernel(
    const float* __restrict__ delta_pre, const float* __restrict__ dt_b,
    const float* __restrict__ xc, const float* __restrict__ dbl,
    const float* __restrict__ A_log, const float* __restrict__ Dp,
    const float* __restrict__ xz, float* __restrict__ yg) {
  int idx = blockIdx.x * 256 + threadIdx.x;
  if (idx >= BSZ * DINNER) return;
  int e = idx & (DINNER - 1);
  int b = idx >> 11;

  float Ae[DSTATE], h[DSTATE];
#pragma unroll
  for (int n = 0; n < DSTATE; ++n) {
    Ae[n] = -__expf(A_log[e * DSTATE + n]);
    h[n] = 0.f;
  }
  const float dtb = dt_b[e];
  const float Dv  = Dp[e];

  for (int l = 0; l < SEQ; ++l) {
    const int row = (b << 10) + l;
    float dp = delta_pre[(size_t)row * DINNER + e] + dtb;
    float delta = (dp > 20.f) ? dp : log1pf(__expf(dp));   // softplus
    float u  = xc[(size_t)row * DINNER + e];
    float du = delta * u;
    const float* bc = dbl + (size_t)row * 128 + DTRANK;    // B_t then C_t
    float y = 0.f;
#pragma unroll
    for (int n = 0; n < DSTATE; ++n) {
      float dA = __expf(delta * Ae[n]);
      h[n] = dA * h[n] + du * bc[n];
      y += h[n] * bc[DSTATE + n];
    }
    float z = xz[(size_t)row * (2 * DINNER) + DINNER + e];
    float g = z / (1.f + __expf(-z));                      // SiLU gate
    yg[(size_t)row * DINNER + e] = (y + u * Dv) * g;
  }
}

// ---------------------------------------------------------------------------
// Host-side orchestration
// ---------------------------------------------------------------------------
extern "C" void kernel_launch(void* const* d_in, const int* in_sizes, int n_in,
                              void* d_out, int out_size, void* d_ws, size_t ws_size,
                              hipStream_t stream) {
  (void)in_sizes; (void)n_in; (void)out_size; (void)ws_size;

  const float* x = (const float*)d_in[0];
  // fwd params: 1..9, bwd params: 10..18, merge_w: 19
  const float* in_proj[2]  = {(const float*)d_in[1],  (const float*)d_in[10]};
  const float* conv_w[2]   = {(const float*)d_in[2],  (const float*)d_in[11]};
  const float* conv_b[2]   = {(const float*)d_in[3],  (const float*)d_in[12]};
  const float* x_proj[2]   = {(const float*)d_in[4],  (const float*)d_in[13]};
  const float* dt_w[2]     = {(const float*)d_in[5],  (const float*)d_in[14]};
  const float* dt_b[2]     = {(const float*)d_in[6],  (const float*)d_in[15]};
  const float* A_log[2]    = {(const float*)d_in[7],  (const float*)d_in[16]};
  const float* Dp[2]       = {(const float*)d_in[8],  (const float*)d_in[17]};
  const float* out_proj[2] = {(const float*)d_in[9],  (const float*)d_in[18]};
  const float* merge_w     = (const float*)d_in[19];
  float* out = (float*)d_out;

  // Workspace carve-out (floats)
  float* ws = (float*)d_ws;
  size_t o = 0;
  float* xrev    = ws + o; o += (size_t)BL * DMODEL;
  float* xz_[2];   xz_[0]   = ws + o; o += (size_t)BL * 2 * DINNER;
                   xz_[1]   = ws + o; o += (size_t)BL * 2 * DINNER;
  float* xc_[2];   xc_[0]   = ws + o; o += (size_t)BL * DINNER;
                   xc_[1]   = ws + o; o += (size_t)BL * DINNER;
  float* dbl_[2];  dbl_[0]  = ws + o; o += (size_t)BL * 128;
                   dbl_[1]  = ws + o; o += (size_t)BL * 128;
  float* dlt_[2];  dlt_[0]  = ws + o; o += (size_t)BL * DINNER;
                   dlt_[1]  = ws + o; o += (size_t)BL * DINNER;
  float* yg_[2];   yg_[0]   = ws + o; o += (size_t)BL * DINNER;
                   yg_[1]   = ws + o; o += (size_t)BL * DINNER;
  float* fb      = ws + o; o += (size_t)BL * 2 * DMODEL;   // [f | flip(b)]
  float* tmpb    = ws + o; o += (size_t)BL * DMODEL;

  dim3 blk(256);

  // 1) xrev = flip_L(x)
  flip_l_kernel<<<dim3((BL * DMODEL + 255) / 256), blk, 0, stream>>>(
      x, xrev, DMODEL, DMODEL, DMODEL, 0);

  for (int d = 0; d < 2; ++d) {
    const float* xin = (d == 0) ? x : xrev;
    // 2) xz = xin @ in_proj^T : (2048 x 1024) * (4096 x 1024)^T
    wmma_gemm_xwt<<<dim3(2 * DINNER / BNW, BL / BM), blk, 0, stream>>>(
        xin, DMODEL, in_proj[d], DMODEL, 2 * DINNER,
        xz_[d], 2 * DINNER, 0, BL, 2 * DINNER, DMODEL);
    // 3) xc = SiLU(conv1d(xi) + b)
    conv_silu_kernel<<<dim3((BL * DINNER + 255) / 256), blk, 0, stream>>>(
        xz_[d], conv_w[d], conv_b[d], xc_[d]);
    // 4) dbl = xc @ x_proj^T : (2048 x 2048) * (96 x 2048)^T  (ldc padded 128)
    wmma_gemm_xwt<<<dim3((96 + BNW - 1) / BNW, BL / BM), blk, 0, stream>>>(
        xc_[d], DINNER, x_proj[d], DINNER, 96,
        dbl_[d], 128, 0, BL, 96, DINNER);
    // 5) delta_pre = dt @ dt_w^T : (2048 x 64) * (2048 x 64)^T
    wmma_gemm_xwt<<<dim3(DINNER / BNW, BL / BM), blk, 0, stream>>>(
        dbl_[d], 128, dt_w[d], DTRANK, DINNER,
        dlt_[d], DINNER, 0, BL, DINNER, DTRANK);
    // 6) selective scan + skip + gate
    scan_kernel<<<dim3((BSZ * DINNER + 255) / 256), blk, 0, stream>>>(
        dlt_[d], dt_b[d], xc_[d], dbl_[d], A_log[d], Dp[d], xz_[d], yg_[d]);
    // 7) out = yg @ out_proj^T : (2048 x 2048) * (1024 x 2048)^T
    float* dst = (d == 0) ? fb : tmpb;
    int   ldc  = (d == 0) ? 2 * DMODEL : DMODEL;
    wmma_gemm_xwt<<<dim3(DMODEL / BNW, BL / BM), blk, 0, stream>>>(
        yg_[d], DINNER, out_proj[d], DINNER, DMODEL,
        dst, ldc, 0, BL, DMODEL, DINNER);
  }

  // 8) fb[:, 1024:2048] = flip_L(bwd out)
  flip_l_kernel<<<dim3((BL * DMODEL + 255) / 256), blk, 0, stream>>>(
      tmpb, fb, DMODEL, DMODEL, 2 * DMODEL, DMODEL);

  // 9) out = fb @ merge_w^T : (2048 x 2048) * (1024 x 2048)^T
  wmma_gemm_xwt<<<dim3(DMODEL / BNW, BL / BM), blk, 0, stream>>>(
      fb, 2 * DMODEL, merge_w, 2 * DMODEL, DMODEL,
      out, DMODEL, 0, BL, DMODEL, 2 * DMODEL);
}